// SAGENet_10419590660556
// MI455X (gfx1250) — compile-verified
//
#include <hip/hip_runtime.h>
#include <hip/hip_bf16.h>

#define N_NODES 50000
#define E_EDGES 800000
#define T_STEPS 8
#define C_FEAT  16              // channels of both layer inputs (C_IN == HID == 16)
#define M_ROWS  (N_NODES * T_STEPS)   // 400000 rows, divisible by 16

typedef __attribute__((ext_vector_type(2))) float v2f;
typedef __attribute__((ext_vector_type(8))) float v8f;

__device__ __forceinline__ void atomAddF(float* p, float v) {
    unsafeAtomicAdd(p, v);   // native global_atomic_add_f32 on gfx1250
}

// ---------------------------------------------------------------------------
// Build H = [ x[res_n_id] | 0 ]  as [M_ROWS, 32] f32.  MODE 0: src = X [T,N,16]
// (layer-1 input layout); MODE 1: src = Y1 [M_ROWS,16] (layer-1 output).
// One thread per float4 of H (M_ROWS*8 threads).
// ---------------------------------------------------------------------------
template <int MODE>
__global__ void init_h_kernel(const float4* __restrict__ src,
                              const int* __restrict__ resid,
                              float4* __restrict__ H) {
    int tid = blockIdx.x * blockDim.x + threadIdx.x;
    if (tid >= M_ROWS * 8) return;
    int m = tid >> 3;
    int j = tid & 7;                    // float4 slot within the 32-float row
    float4 v = make_float4(0.f, 0.f, 0.f, 0.f);
    if (j < 4) {                        // residual half
        int n = m / T_STEPS, t = m % T_STEPS;
        int rn = resid[n];
        if (MODE == 0) v = src[(t * N_NODES + rn) * 4 + j];
        else           v = src[(rn * T_STEPS + t) * 4 + j];
    }
    H[tid] = v;
}

__global__ void zero_cnt_kernel(float* __restrict__ cnt) {
    int i = blockIdx.x * blockDim.x + threadIdx.x;
    if (i < N_NODES) cnt[i] = 0.f;
}

__global__ void count_deg_kernel(const int* __restrict__ ei, float* __restrict__ cnt) {
    int e = blockIdx.x * blockDim.x + threadIdx.x;
    if (e < E_EDGES) atomAddF(&cnt[ei[E_EDGES + e]], 1.f);
}

// ---------------------------------------------------------------------------
// Edge scatter: one wave32 per edge. Lane l covers (t = l>>2, chans 4*(l&3)..+3).
// MODE 0: feat = X [T,N,16];  MODE 1: feat = Y1 [N,T,16] (m-indexed).
// Accumulates w * x[src] into the agg half of H (columns 16..31).
// ---------------------------------------------------------------------------
template <int MODE>
__global__ void scatter_kernel(const int* __restrict__ ei,
                               const float* __restrict__ ew,
                               const float4* __restrict__ feat,
                               float* __restrict__ H) {
    unsigned tid = blockIdx.x * blockDim.x + threadIdx.x;
    int e = (int)(tid >> 5);
    int l = (int)(tid & 31);
    if (e >= E_EDGES) return;
    int s = ei[e];
    int d = ei[E_EDGES + e];
    float w = ew[e];
    int t = l >> 2, q = l & 3;
    float4 x;
    if (MODE == 0) x = feat[(t * N_NODES + s) * 4 + q];
    else           x = feat[(s * T_STEPS + t) * 4 + q];
    float* hp = H + (d * T_STEPS + t) * 32 + 16 + q * 4;
    atomAddF(hp + 0, w * x.x);
    atomAddF(hp + 1, w * x.y);
    atomAddF(hp + 2, w * x.z);
    atomAddF(hp + 3, w * x.w);
}

// ---------------------------------------------------------------------------
// Fused mean / GEMM / bias / leaky-ReLU using V_WMMA_F32_16X16X4_F32.
// One wave per 16-row tile of H [M_ROWS, 32]; K=32 via 8 WMMA steps.
// A frag (16x4 f32): lane&15 -> row, lanes>=16 take K=k0+2..3 (per ISA layout).
// Mean normalization (1/max(cnt,1)) fused into agg-half A loads.
// OUTMODE 0: out = Y1[m*16 + c].  OUTMODE 1: out[t*N*32 + n*32 + c] ([T,N,32]).
// ---------------------------------------------------------------------------
template <int COUT, int OUTMODE>
__global__ void gemm_wmma_kernel(const float* __restrict__ H,
                                 const float* __restrict__ cnt,
                                 const float* __restrict__ W,
                                 const float* __restrict__ bias,
                                 float* __restrict__ out) {
    int wave = (blockIdx.x * blockDim.x + threadIdx.x) >> 5;
    int lane = threadIdx.x & 31;
    if (wave >= M_ROWS / 16) return;          // never taken (exact grid)
    const int rlo   = lane & 15;
    const int khalf = lane >> 4;              // 0: K=k0,k0+1  1: K=k0+2,k0+3
    const int row   = wave * 16 + rlo;
    const float invc = 1.0f / fmaxf(cnt[row / T_STEPS], 1.0f);
    const float* hrow = H + row * 32;

    constexpr int NT = COUT / 16;
    v8f acc[NT];
#pragma unroll
    for (int nt = 0; nt < NT; ++nt) acc[nt] = (v8f){};

#pragma unroll
    for (int k0 = 0; k0 < 32; k0 += 4) {
        const int ka = k0 + 2 * khalf;
        v2f a;
        a.x = hrow[ka];
        a.y = hrow[ka + 1];
        if (ka >= 16) { a.x *= invc; a.y *= invc; }   // agg half -> mean
#pragma unroll
        for (int nt = 0; nt < NT; ++nt) {
            v2f b;
            b.x = W[ka * COUT + nt * 16 + rlo];
            b.y = W[(ka + 1) * COUT + nt * 16 + rlo];
            acc[nt] = __builtin_amdgcn_wmma_f32_16x16x4_f32(
                false, a, false, b, (short)0, acc[nt], false, false);
        }
    }

#pragma unroll
    for (int nt = 0; nt < NT; ++nt) {
        const int col = nt * 16 + rlo;
        const float bv = bias[col];
#pragma unroll
        for (int v = 0; v < 8; ++v) {
            const int m = wave * 16 + v + 8 * khalf;  // D layout: M = v + 8*hi
            float val = acc[nt][v] + bv;
            val = val > 0.f ? val : 0.01f * val;      // leaky_relu(0.01)
            if (OUTMODE == 0) {
                out[m * 16 + col] = val;
            } else {
                const int n = m / T_STEPS, t = m % T_STEPS;
                out[(t * N_NODES + n) * 32 + col] = val;
            }
        }
    }
}

// ---------------------------------------------------------------------------
extern "C" void kernel_launch(void* const* d_in, const int* in_sizes, int n_in,
                              void* d_out, int out_size, void* d_ws, size_t ws_size,
                              hipStream_t stream) {
    const float* X   = (const float*)d_in[0];
    const int*   ei0 = (const int*)  d_in[1];
    const float* ew0 = (const float*)d_in[2];
    const int*   ei1 = (const int*)  d_in[3];
    const float* ew1 = (const float*)d_in[4];
    const int*   rn0 = (const int*)  d_in[5];
    const int*   rn1 = (const int*)  d_in[6];
    const float* W1  = (const float*)d_in[7];
    const float* b1  = (const float*)d_in[8];
    const float* W2  = (const float*)d_in[9];
    const float* b2  = (const float*)d_in[10];
    float* out = (float*)d_out;

    // Workspace layout (floats): cnt | Y1 [M,16] | H [M,32]   (~77 MB total)
    float* ws  = (float*)d_ws;
    float* cnt = ws;
    float* Y1  = ws + 65536;
    float* H   = Y1 + (size_t)M_ROWS * 16;

    const int BLK = 256;
    const int gInit    = (M_ROWS * 8) / BLK;         // 12500
    const int gCnt0    = (N_NODES + BLK - 1) / BLK;  // 196
    const int gDeg     = (E_EDGES + BLK - 1) / BLK;  // 3125
    const int gScatter = (E_EDGES * 32) / BLK;       // 100000
    const int gGemm    = (M_ROWS / 16) / 8;          // 3125 (8 waves / block)

    // ---- Layer 1: X [T,N,16] -> Y1 [M,16] ----
    init_h_kernel<0><<<gInit, BLK, 0, stream>>>((const float4*)X, rn0, (float4*)H);
    zero_cnt_kernel<<<gCnt0, BLK, 0, stream>>>(cnt);
    count_deg_kernel<<<gDeg, BLK, 0, stream>>>(ei0, cnt);
    scatter_kernel<0><<<gScatter, BLK, 0, stream>>>(ei0, ew0, (const float4*)X, H);
    gemm_wmma_kernel<16, 0><<<gGemm, BLK, 0, stream>>>(H, cnt, W1, b1, Y1);

    // ---- Layer 2: Y1 [M,16] -> out [T,N,32] ----
    init_h_kernel<1><<<gInit, BLK, 0, stream>>>((const float4*)Y1, rn1, (float4*)H);
    zero_cnt_kernel<<<gCnt0, BLK, 0, stream>>>(cnt);
    count_deg_kernel<<<gDeg, BLK, 0, stream>>>(ei1, cnt);
    scatter_kernel<1><<<gScatter, BLK, 0, stream>>>(ei1, ew1, (const float4*)Y1, H);
    gemm_wmma_kernel<32, 1><<<gGemm, BLK, 0, stream>>>(H, cnt, W2, b2, out);
}